// Router_51281909514476
// MI455X (gfx1250) — compile-verified
//
#include <hip/hip_runtime.h>

// ---------------------------------------------------------------------------
// MoE router for MI455X (gfx1250, wave32).
// GEMM via V_WMMA_F32_16X16X4_F32 (f32 in/out, matches reference numerics).
// Memory-bound: 134 MB of x @ 23.3 TB/s ~= 5.8us floor; W (512KB) stays in L2.
// One wave = 16 tokens x 64 experts (4 accumulator tiles of 16x16 f32).
// ---------------------------------------------------------------------------

typedef __attribute__((ext_vector_type(2))) float v2f;
typedef __attribute__((ext_vector_type(8))) float v8f;

#define TOKENS  16384
#define DMODEL  2048
#define NEXP    64
#define WPB     8           // waves per block
#define TPW     16          // tokens per wave (M tile)

__global__ __launch_bounds__(256)
void router_gemm_softmax_top2(const float* __restrict__ x,
                              const float* __restrict__ W,
                              const float* __restrict__ bias,
                              float* __restrict__ out,      // [0,32768) idx, [32768,65536) scores
                              float* __restrict__ cnt,      // 64 floats (dispatch counts)
                              float* __restrict__ psumg)    // 64 floats (prob sums)
{
    const int lane = threadIdx.x & 31;
    const int wave = threadIdx.x >> 5;
    const int half = lane >> 4;     // 0: rows m=0..7 / K lo,  1: rows m=8..15 / K hi
    const int li   = lane & 15;     // A: row m ; B/C/D: col n
    const int token_base = (blockIdx.x * WPB + wave) * TPW;

    v8f acc0 = {}, acc1 = {}, acc2 = {}, acc3 = {};

    // A: lane holds x[token_base+li][k] for k = kk + 2*half + {0,1}
    const float* xrow = x + (size_t)(token_base + li) * DMODEL + 2 * half;
    // B: lane holds W[kk + 2*half + v][e0 + li], v = 0..1
    const float* wcol = W + (size_t)(2 * half) * NEXP + li;

    #pragma unroll 4
    for (int kk = 0; kk < DMODEL; kk += 4) {
        v2f a = *(const v2f*)(xrow + kk);
        const float* wp = wcol + (size_t)kk * NEXP;
        v2f b0, b1, b2, b3;
        b0[0] = wp[0];  b0[1] = wp[NEXP + 0];
        b1[0] = wp[16]; b1[1] = wp[NEXP + 16];
        b2[0] = wp[32]; b2[1] = wp[NEXP + 32];
        b3[0] = wp[48]; b3[1] = wp[NEXP + 48];
        acc0 = __builtin_amdgcn_wmma_f32_16x16x4_f32(false, a, false, b0, (short)0, acc0, false, false);
        acc1 = __builtin_amdgcn_wmma_f32_16x16x4_f32(false, a, false, b1, (short)0, acc1, false, false);
        acc2 = __builtin_amdgcn_wmma_f32_16x16x4_f32(false, a, false, b2, (short)0, acc2, false, false);
        acc3 = __builtin_amdgcn_wmma_f32_16x16x4_f32(false, a, false, b3, (short)0, acc3, false, false);
    }

    // Bias (depends on expert column n = li only)
    const float bv0 = bias[li], bv1 = bias[16 + li], bv2 = bias[32 + li], bv3 = bias[48 + li];
    #pragma unroll
    for (int v = 0; v < 8; ++v) {
        acc0[v] += bv0; acc1[v] += bv1; acc2[v] += bv2; acc3[v] += bv3;
    }

    float psum_part0 = 0.f, psum_part1 = 0.f, psum_part2 = 0.f, psum_part3 = 0.f;

    // Per accumulator row v: token t = token_base + v + 8*half, logits across 64 experts
    // spread over 16 lanes x 4 tiles.  xor-shuffles 1,2,4,8 stay inside each half.
    #pragma unroll
    for (int v = 0; v < 8; ++v) {
        float l0 = acc0[v], l1 = acc1[v], l2 = acc2[v], l3 = acc3[v];

        float mx = fmaxf(fmaxf(l0, l1), fmaxf(l2, l3));
        #pragma unroll
        for (int d = 1; d <= 8; d <<= 1) mx = fmaxf(mx, __shfl_xor(mx, d, 32));

        float e0 = __expf(l0 - mx), e1 = __expf(l1 - mx);
        float e2 = __expf(l2 - mx), e3 = __expf(l3 - mx);
        float s = e0 + e1 + e2 + e3;
        #pragma unroll
        for (int d = 1; d <= 8; d <<= 1) s += __shfl_xor(s, d, 32);
        const float inv = 1.0f / s;
        const float p0 = e0 * inv, p1 = e1 * inv, p2 = e2 * inv, p3 = e3 * inv;

        psum_part0 += p0; psum_part1 += p1; psum_part2 += p2; psum_part3 += p3;

        // local top-2 among this lane's 4 experts
        float a1 = p0, a2 = p1; int j1 = li, j2 = 16 + li;
        if (a2 > a1) { float tf = a1; a1 = a2; a2 = tf; int ti = j1; j1 = j2; j2 = ti; }
        if (p2 > a1)      { a2 = a1; j2 = j1; a1 = p2; j1 = 32 + li; }
        else if (p2 > a2) { a2 = p2; j2 = 32 + li; }
        if (p3 > a1)      { a2 = a1; j2 = j1; a1 = p3; j1 = 48 + li; }
        else if (p3 > a2) { a2 = p3; j2 = 48 + li; }

        // merge top-2 pairs across the 16 lanes of this half
        #pragma unroll
        for (int d = 1; d <= 8; d <<= 1) {
            float o1 = __shfl_xor(a1, d, 32); int oj1 = __shfl_xor(j1, d, 32);
            float o2 = __shfl_xor(a2, d, 32); int oj2 = __shfl_xor(j2, d, 32);
            if (o1 > a1) {
                if (a1 >= o2) { a2 = a1; j2 = j1; } else { a2 = o2; j2 = oj2; }
                a1 = o1; j1 = oj1;
            } else if (o1 > a2) {
                a2 = o1; j2 = oj1;
            }
        }

        if (li == 0) {
            const int t = token_base + v + 8 * half;
            out[2 * t + 0] = (float)j1;
            out[2 * t + 1] = (float)j2;
            out[2 * TOKENS + 2 * t + 0] = a1;
            out[2 * TOKENS + 2 * t + 1] = a2;
            atomicAdd(&cnt[j1], 1.0f);
            atomicAdd(&cnt[j2], 1.0f);
        }
    }

    // fold the two halves' prob partial sums, then one atomic per expert per wave
    psum_part0 += __shfl_xor(psum_part0, 16, 32);
    psum_part1 += __shfl_xor(psum_part1, 16, 32);
    psum_part2 += __shfl_xor(psum_part2, 16, 32);
    psum_part3 += __shfl_xor(psum_part3, 16, 32);
    if (half == 0) {
        atomicAdd(&psumg[li],      psum_part0);
        atomicAdd(&psumg[16 + li], psum_part1);
        atomicAdd(&psumg[32 + li], psum_part2);
        atomicAdd(&psumg[48 + li], psum_part3);
    }
}

__global__ void router_zero_ws(float* ws)
{
    ws[threadIdx.x] = 0.0f;   // 128 floats: cnt[64] + psum[64]
}

__global__ void router_finalize_aux(const float* __restrict__ cnt,
                                    const float* __restrict__ psumg,
                                    float* __restrict__ out_aux)
{
    const int l = threadIdx.x;      // 32 threads, one wave
    float prod = cnt[l] * psumg[l] + cnt[l + 32] * psumg[l + 32];
    #pragma unroll
    for (int d = 1; d <= 16; d <<= 1) prod += __shfl_xor(prod, d, 32);
    if (l == 0) {
        // aux = E * sum_e (cnt_e / N) * (psum_e / N),  N = 16384, E = 64
        *out_aux = 64.0f * prod * (1.0f / (16384.0f * 16384.0f));
    }
}

extern "C" void kernel_launch(void* const* d_in, const int* in_sizes, int n_in,
                              void* d_out, int out_size, void* d_ws, size_t ws_size,
                              hipStream_t stream)
{
    const float* x    = (const float*)d_in[0];   // (4,4096,2048) f32
    const float* W    = (const float*)d_in[1];   // (2048,64) f32
    const float* bias = (const float*)d_in[2];   // (64,) f32
    float* out = (float*)d_out;                  // 32768 idx + 32768 scores + 1 aux
    float* ws  = (float*)d_ws;                   // cnt[64] | psum[64]

    router_zero_ws<<<1, 128, 0, stream>>>(ws);
    router_gemm_softmax_top2<<<TOKENS / (WPB * TPW), 32 * WPB, 0, stream>>>(
        x, W, bias, out, ws, ws + 64);
    router_finalize_aux<<<1, 32, 0, stream>>>(ws, ws + 64, out + 2 * 2 * TOKENS);
}